// MappedAvgPool_34282428956673
// MI455X (gfx1250) — compile-verified
//
#include <hip/hip_runtime.h>
#include <hip/hip_bf16.h>
#include <stdint.h>

// ---------------------------------------------------------------------------
// MappedAvgPool for MI455X (gfx1250)
//   x:        (2,32,512,512) f32
//   map:      (256,256,9,4,2) i32   (x-coord, y-coord)
//   weights:  (256,256,9,4)  f32
//   out:      (2,32,256,256) f32 = sum_{k,p} x[b,c,y,x]*w / 9
// Strategy: transpose x to channels-last (one 128B line per sample), stage
// per-pixel tables via CDNA5 async global->LDS loads, coalesced everything.
// ---------------------------------------------------------------------------

#define B_  2
#define C_  32
#define H_  512
#define W_  512
#define OH_ 256
#define OW_ 256
#define KP_ 36            // K*P = 9*4
#define HW_ (H_ * W_)     // 262144

// ---- CDNA5 async global->LDS helpers --------------------------------------
#if defined(__has_builtin)
#  if __has_builtin(__builtin_amdgcn_global_load_async_to_lds_b32)
#    define HAVE_ASYNC_LDS_BUILTIN 1
#  endif
#  if __has_builtin(__builtin_amdgcn_s_wait_asynccnt)
#    define HAVE_WAIT_ASYNC_BUILTIN 1
#  endif
#endif

typedef __attribute__((address_space(1))) int g_int;
typedef __attribute__((address_space(3))) int l_int;

__device__ __forceinline__ void async_g2l_b32(void* lds_dst, const void* gsrc) {
#if defined(HAVE_ASYNC_LDS_BUILTIN)
  void* g = const_cast<void*>(gsrc);
  __builtin_amdgcn_global_load_async_to_lds_b32(
      (g_int*)g, (l_int*)lds_dst, /*imm offset*/ 0, /*cpol*/ 0);
#else
  unsigned lds_off = (unsigned)(uintptr_t)lds_dst;
  asm volatile("global_load_async_to_lds_b32 %0, %1, off"
               :: "v"(lds_off), "v"(gsrc) : "memory");
#endif
}

__device__ __forceinline__ void wait_async0() {
#if defined(HAVE_WAIT_ASYNC_BUILTIN)
  __builtin_amdgcn_s_wait_asynccnt(0);
#else
  asm volatile("s_wait_asynccnt 0" ::: "memory");
#endif
}

// ---------------------------------------------------------------------------
// Kernel 1: x(B,C,H,W) -> xt(B,H,W,C).  32x32 LDS tile transpose, fully
// coalesced on both sides.  grid = (W/32, H, B), block = 256.
// ---------------------------------------------------------------------------
__global__ __launch_bounds__(256) void mapool_transpose(
    const float* __restrict__ x, float* __restrict__ xt) {
  __shared__ float t[32][33];
  const int tx = threadIdx.x & 31;
  const int ty = threadIdx.x >> 5;          // 0..7
  const int x0 = blockIdx.x << 5;           // w tile origin
  const int y  = blockIdx.y;
  const int b  = blockIdx.z;

#pragma unroll
  for (int i = 0; i < 4; ++i) {
    const int c = ty + (i << 3);
    t[c][tx] = x[(((size_t)(b * C_ + c) * H_ + y) * W_) + x0 + tx];
  }
  __syncthreads();
#pragma unroll
  for (int i = 0; i < 4; ++i) {
    const int j = ty + (i << 3);            // w within tile
    xt[(((size_t)(b * H_ + y) * W_) + x0 + j) * C_ + tx] = t[tx][j];
  }
}

// ---------------------------------------------------------------------------
// Kernel 2: gather + weighted reduce.
// grid = (OW/32, OH, B), block = 256 (8 wave32).
// Block handles 32 consecutive v-pixels of row u for batch b:
//   - async-stage 32*36 (x,y) pairs + weights into LDS
//   - transform to byte offsets into xt (plane folded in), w *= 1/9
//   - wave w computes pixels {w, w+8, w+16, w+24}; lane = channel.
//     Each sample read = one contiguous 128B cacheline (L2-resident xt).
//   - transpose result tile in LDS -> coalesced (B,C,OH,OW) stores.
// ---------------------------------------------------------------------------
__global__ __launch_bounds__(256) void mapool_gather(
    const int*   __restrict__ smap,
    const float* __restrict__ wgt,
    const float* __restrict__ xt,
    float*       __restrict__ out) {
  __shared__ int      raw_map[32 * KP_ * 2];   // 9216 B
  __shared__ float    raw_w [32 * KP_];        // 4608 B
  __shared__ unsigned s_off [32 * KP_];        // byte offsets into xt
  __shared__ float    s_w   [32 * KP_];
  __shared__ float    tile[32][33];            // [pixel][channel]

  const int tid = threadIdx.x;
  const int v0  = blockIdx.x << 5;
  const int u   = blockIdx.y;
  const int b   = blockIdx.z;
  const int pixbase = u * OW_ + v0;            // first pixel of strip

  // ---- Stage 1: async copy raw tables to LDS (ASYNCcnt path) ----
  const int*   gmap = smap + (size_t)pixbase * (KP_ * 2);
  const float* gw   = wgt  + (size_t)pixbase * KP_;
  for (int i = tid; i < 32 * KP_ * 2; i += 256)
    async_g2l_b32(&raw_map[i], gmap + i);
  for (int i = tid; i < 32 * KP_; i += 256)
    async_g2l_b32(&raw_w[i], gw + i);
  wait_async0();
  __syncthreads();

  // ---- Stage 2: transform (x,y) -> byte offset, fold 1/K into weight ----
  const unsigned planeBytes = (unsigned)b * (unsigned)(HW_ * C_ * 4);
  for (int i = tid; i < 32 * KP_; i += 256) {
    const int xi = raw_map[2 * i + 0];
    const int yi = raw_map[2 * i + 1];
    s_off[i] = planeBytes + (unsigned)(yi * W_ + xi) * (unsigned)(C_ * 4);
    s_w[i]   = raw_w[i] * (1.0f / 9.0f);
  }
  __syncthreads();

  // ---- Stage 3: gather + reduce (lane = channel, 128B line per sample) ----
  const int lane = tid & 31;
  const int wv   = tid >> 5;                  // 0..7
  const char* base = (const char*)xt;
  const int p0 = wv, p1 = wv + 8, p2 = wv + 16, p3 = wv + 24;
  float a0 = 0.f, a1 = 0.f, a2 = 0.f, a3 = 0.f;
#pragma unroll 4
  for (int j = 0; j < KP_; ++j) {
    const float x0f = ((const float*)(base + s_off[p0 * KP_ + j]))[lane];
    const float x1f = ((const float*)(base + s_off[p1 * KP_ + j]))[lane];
    const float x2f = ((const float*)(base + s_off[p2 * KP_ + j]))[lane];
    const float x3f = ((const float*)(base + s_off[p3 * KP_ + j]))[lane];
    a0 = fmaf(x0f, s_w[p0 * KP_ + j], a0);
    a1 = fmaf(x1f, s_w[p1 * KP_ + j], a1);
    a2 = fmaf(x2f, s_w[p2 * KP_ + j], a2);
    a3 = fmaf(x3f, s_w[p3 * KP_ + j], a3);
  }
  tile[p0][lane] = a0;
  tile[p1][lane] = a1;
  tile[p2][lane] = a2;
  tile[p3][lane] = a3;
  __syncthreads();

  // ---- Stage 4: coalesced stores to (B,C,OH,OW) ----
#pragma unroll
  for (int i = 0; i < 4; ++i) {
    const int c = wv + (i << 3);
    out[(((size_t)(b * C_ + c) * OH_ + u) * OW_) + v0 + lane] = tile[lane][c];
  }
}

// ---------------------------------------------------------------------------
// Fallback (workspace too small): one thread per output element.
// ---------------------------------------------------------------------------
__global__ __launch_bounds__(256) void mapool_naive(
    const float* __restrict__ x, const int* __restrict__ smap,
    const float* __restrict__ wgt, float* __restrict__ out) {
  const int idx = blockIdx.x * 256 + threadIdx.x;
  if (idx >= B_ * C_ * OH_ * OW_) return;
  const int v = idx & (OW_ - 1);
  const int u = (idx >> 8) & (OH_ - 1);
  const int c = (idx >> 16) & (C_ - 1);
  const int b = idx >> 21;
  const int*   m  = smap + (size_t)(u * OW_ + v) * (KP_ * 2);
  const float* wp = wgt  + (size_t)(u * OW_ + v) * KP_;
  const float* plane = x + (size_t)(b * C_ + c) * HW_;
  float acc = 0.f;
#pragma unroll 4
  for (int j = 0; j < KP_; ++j) {
    const int xi = m[2 * j], yi = m[2 * j + 1];
    acc = fmaf(plane[yi * W_ + xi], wp[j], acc);
  }
  out[idx] = acc * (1.0f / 9.0f);
}

// ---------------------------------------------------------------------------
extern "C" void kernel_launch(void* const* d_in, const int* in_sizes, int n_in,
                              void* d_out, int out_size, void* d_ws, size_t ws_size,
                              hipStream_t stream) {
  const float* x    = (const float*)d_in[0];
  const int*   smap = (const int*)d_in[1];
  const float* wgt  = (const float*)d_in[2];
  float*       out  = (float*)d_out;

  const size_t xtBytes = (size_t)B_ * H_ * W_ * C_ * sizeof(float); // 64 MB
  if (ws_size >= xtBytes) {
    float* xt = (float*)d_ws;
    mapool_transpose<<<dim3(W_ / 32, H_, B_), 256, 0, stream>>>(x, xt);
    mapool_gather<<<dim3(OW_ / 32, OH_, B_), 256, 0, stream>>>(smap, wgt, xt, out);
  } else {
    const int n = B_ * C_ * OH_ * OW_;
    mapool_naive<<<(n + 255) / 256, 256, 0, stream>>>(x, smap, wgt, out);
  }
}